// CutStripes_29523605193347
// MI455X (gfx1250) — compile-verified
//
#include <hip/hip_runtime.h>

// ---------------------------------------------------------------------------
// CutStripes: out[b,0,t,f] = mask(b,t) ? x[perm[b],0,t,f] : x[b,0,t,f]
// Zero-FLOP data movement; roofline = 256 MB / 23.3 TB/s ~= 11 us.
// CDNA5 path: 16-deep async global->LDS b128 pipeline (ASYNCcnt) per wave,
// one s_wait_asynccnt drain, then ds_load_b128 -> NT global_store_b128.
// ---------------------------------------------------------------------------

#define B_DIM 128
#define T_DIM 2048
#define F_DIM 128
#define CHUNK 16           // rows staged per wave (async pipeline depth)
#define WAVES 8            // waves per block (256 threads, wave32)

typedef float v4f __attribute__((ext_vector_type(4)));
typedef int   v4i __attribute__((ext_vector_type(4)));
typedef __attribute__((address_space(1))) v4i gv4i_t;   // global 128-bit chunk
typedef __attribute__((address_space(3))) v4i lv4i_t;   // LDS    128-bit chunk

__device__ __forceinline__ void async_copy_b128(const float* gsrc, float* ldst) {
#if __has_builtin(__builtin_amdgcn_global_load_async_to_lds_b128)
  __builtin_amdgcn_global_load_async_to_lds_b128(
      (gv4i_t*)gsrc, (lv4i_t*)ldst, /*offset=*/0, /*cpol=*/0);
#else
  unsigned lofs = (unsigned)(unsigned long long)(lv4i_t*)ldst;
  unsigned long long gaddr = (unsigned long long)gsrc;
  asm volatile("global_load_async_to_lds_b128 %0, %1, off"
               :: "v"(lofs), "v"(gaddr) : "memory");
#endif
}

__device__ __forceinline__ void wait_async0() {
#if __has_builtin(__builtin_amdgcn_s_wait_asynccnt)
  __builtin_amdgcn_s_wait_asynccnt(0);
#else
  asm volatile("s_wait_asynccnt 0" ::: "memory");
#endif
  asm volatile("" ::: "memory");   // compiler barrier: keep LDS reads below
}

__global__ __launch_bounds__(256) void CutStripes_kernel(
    const float* __restrict__ x, const int* __restrict__ perm,
    const int* __restrict__ bgn, const int* __restrict__ dist,
    float* __restrict__ out)
{
  // 8 waves * 16 rows * 512 B = 64 KB staging (LDS pool is 320 KB/WGP)
  __shared__ float smem[WAVES * CHUNK * F_DIM];

  // 16 blocks per batch -> b is block-uniform -> metadata via scalar loads
  const int b  = (int)blockIdx.x >> 4;
  const int t0 = ((int)blockIdx.x & 15) << 7;     // 128 rows per block
  // wave id is wave-uniform: pin it to an SGPR so mask math stays scalar
  const int wv = __builtin_amdgcn_readfirstlane((int)threadIdx.x >> 5);
  const int ln = (int)threadIdx.x & 31;

  const int pb  = perm[b];
  const int lo0 = bgn[b * 4 + 0], lo1 = bgn[b * 4 + 1];
  const int lo2 = bgn[b * 4 + 2], lo3 = bgn[b * 4 + 3];
  const int hi0 = lo0 + dist[b * 4 + 0], hi1 = lo1 + dist[b * 4 + 1];
  const int hi2 = lo2 + dist[b * 4 + 2], hi3 = lo3 + dist[b * 4 + 3];

  const float* __restrict__ xb = x   + (size_t)b  * (T_DIM * F_DIM);
  const float* __restrict__ xp = x   + (size_t)pb * (T_DIM * F_DIM);
  float*       __restrict__ ob = out + (size_t)b  * (T_DIM * F_DIM);

  float* wbuf = smem + wv * (CHUNK * F_DIM);      // wave-private staging
  const int lane_off = ln * 4;                    // float offset of lane's b128
  const int tb = t0 + wv * CHUNK;                 // wave-uniform row base

  // ---- Fire 16 async b128 loads (mask-selected source); no VGPR data path.
  #pragma unroll
  for (int i = 0; i < CHUNK; ++i) {
    const int t = tb + i;                         // wave-uniform (SGPR)
    const bool m = (t >= lo0 && t < hi0) || (t >= lo1 && t < hi1) ||
                   (t >= lo2 && t < hi2) || (t >= lo3 && t < hi3);
    const float* src = (m ? xp : xb) + (size_t)t * F_DIM + lane_off;
    async_copy_b128(src, wbuf + i * F_DIM + lane_off);
  }

  wait_async0();   // single drain; ASYNCcnt is loads-only here -> in order

  // ---- LDS -> VGPR -> global (NT: output never re-read; spare the L2)
  #pragma unroll
  for (int i = 0; i < CHUNK; ++i) {
    v4f v = *reinterpret_cast<const v4f*>(wbuf + i * F_DIM + lane_off);
    __builtin_nontemporal_store(
        v, reinterpret_cast<v4f*>(ob + (size_t)(tb + i) * F_DIM + lane_off));
  }
  // Stores are fire-and-forget (STOREcnt); s_endpgm performs the final wait.
}

extern "C" void kernel_launch(void* const* d_in, const int* in_sizes, int n_in,
                              void* d_out, int out_size, void* d_ws, size_t ws_size,
                              hipStream_t stream) {
  (void)in_sizes; (void)n_in; (void)out_size; (void)d_ws; (void)ws_size;
  const float* x    = (const float*)d_in[0];
  const int*   perm = (const int*)d_in[1];
  const int*   bgn  = (const int*)d_in[2];
  const int*   dist = (const int*)d_in[3];
  float*       out  = (float*)d_out;

  dim3 grid(B_DIM * 16);   // 2048 blocks, each: one batch b, 128 t-rows
  dim3 block(256);         // 8 wave32
  CutStripes_kernel<<<grid, block, 0, stream>>>(x, perm, bgn, dist, out);
}